// MultiHeadAttentionQuantum_65481071410268
// MI455X (gfx1250) — compile-verified
//
#include <hip/hip_runtime.h>
#include <math.h>

typedef __attribute__((ext_vector_type(2))) float v2f;
typedef __attribute__((ext_vector_type(8))) float v8f;

#define NW 8
#define BB 8
#define SS 1024
#define EE 128
#define HH 16
#define DK 8
#define ZS 18   // sZ row stride (floats): 8 z + 1 one + 7 zeros + 2 pad

// ---------------------------------------------------------------------------
// Kernel 1 (fused): quantum encoding + self-attention, one block per (b, h).
// z never touches HBM. Scores are bounded (|s| <= sqrt(8)), so softmax needs
// no max subtraction: P = exp(s), O = sum P@V, L = sum rowsum(P).
// sZ rows are materialized as [z0..z7 | 1 | 0 0 0 0 0 0 0 | pad pad] so the
// PV B-operand (16 columns: V, ones-column for L, zero padding) is a plain
// unconditional LDS load -- no exec-mask predication in the inner loop.
//   sZ: [1024][18] f32, 72KB   sP: per-wave 16x18 f32 P-restage scratch, 9KB
// ---------------------------------------------------------------------------
__global__ __launch_bounds__(256)
void qattn_kernel(const float* __restrict__ x, const float* __restrict__ theta,
                  float* __restrict__ y) {
  __shared__ float sZ[SS * ZS];
  __shared__ float sP[8 * 16 * 18];

  const int bh  = blockIdx.x;          // 0..127
  const int b   = bh / HH;
  const int h   = bh % HH;
  const int tid = threadIdx.x;

  float th[NW];
#pragma unroll
  for (int j = 0; j < NW; ++j) th[j] = theta[j];

  // --- quantum encode: z_w = prefix/suffix products of cos(x_j + theta_j) ---
  for (int s = tid; s < SS; s += 256) {
    const float* xr = x + ((size_t)(b * SS + s) * EE + h * NW);
    float c[NW];
#pragma unroll
    for (int j = 0; j < NW; ++j) c[j] = __cosf(xr[j] + th[j]);
    float zr[NW];
    float p = c[0];
#pragma unroll
    for (int j = 1; j < NW; ++j) { p *= c[j]; zr[j] = p; }
    float sfx = 1.0f;
#pragma unroll
    for (int j = NW - 1; j >= 1; --j) sfx *= c[j];
    zr[0] = sfx;
#pragma unroll
    for (int j = 0; j < NW; ++j) sZ[s * ZS + j] = zr[j];
    sZ[s * ZS + 8] = 1.0f;               // ones column -> accumulates L
#pragma unroll
    for (int j = 9; j < 16; ++j) sZ[s * ZS + j] = 0.0f;  // zero padding
  }
  __syncthreads();

  // --- attention: one wave per 16-query block (8 blocks per wave) ---
  const int lane = tid & 31;
  const int wave = tid >> 5;
  const int hf   = lane >> 4;   // lane half (0: lanes 0-15, 1: lanes 16-31)
  const int lm   = lane & 15;
  const float scale = 0.35355339059327373f;  // 1/sqrt(8)
  float* sp = sP + wave * (16 * 18);

  for (int qb = wave; qb < SS / 16; qb += 8) {
    const int q0 = qb * 16;

    // A operands: Q tile 16x8, pre-scaled (A layout: m=lm, k = 2*hf + {0,1})
    const float* qp = &sZ[(q0 + lm) * ZS + hf * 2];
    v2f a0 = {qp[0] * scale, qp[1] * scale};
    v2f a1 = {qp[4] * scale, qp[5] * scale};

    // O accumulator: cols 0..7 = P@V, col 8 = L (rowsum of P)
    v8f o = {0.f, 0.f, 0.f, 0.f, 0.f, 0.f, 0.f, 0.f};

    for (int kb = 0; kb < SS / 16; ++kb) {
      const int k0 = kb * 16;

      // B operands: K^T chunks (n=lm, k = 2*hf + {0,1})
      const float* kp = &sZ[(k0 + lm) * ZS + hf * 2];
      v2f b0 = {kp[0], kp[1]};
      v2f b1 = {kp[4], kp[5]};

      v8f sc = {0.f, 0.f, 0.f, 0.f, 0.f, 0.f, 0.f, 0.f};
      sc = __builtin_amdgcn_wmma_f32_16x16x4_f32(false, a0, false, b0,
                                                 (short)0, sc, false, false);
      sc = __builtin_amdgcn_wmma_f32_16x16x4_f32(false, a1, false, b1,
                                                 (short)0, sc, false, false);

      // P = exp(scores): bounded, no max subtraction needed
      float pr[8];
#pragma unroll
      for (int r = 0; r < 8; ++r) pr[r] = __expf(sc[r]);

      // stage P (16x16) to LDS so it can be re-read in A layout
#pragma unroll
      for (int r = 0; r < 8; ++r)
        sp[(r + 8 * hf) * 18 + lm] = pr[r];
      __builtin_amdgcn_wave_barrier();

      // O += P @ [V | 1 | 0...] : K=16 -> 4 chained f32 WMMAs.
      // B element (k, n=lm) = sZ[(k0+k)*ZS + lm]; cols 8..15 already hold
      // the ones/zeros, so these loads are unconditional.
#pragma unroll
      for (int c = 0; c < 4; ++c) {
        const float* pp = &sp[lm * 18 + c * 4 + hf * 2];
        v2f pa = {pp[0], pp[1]};
        const int krow = k0 + c * 4 + hf * 2;
        v2f vb = {sZ[krow * ZS + lm], sZ[(krow + 1) * ZS + lm]};
        o = __builtin_amdgcn_wmma_f32_16x16x4_f32(false, pa, false, vb,
                                                  (short)0, o, false, false);
      }
      __builtin_amdgcn_wave_barrier();
    }

    // normalize: L for row (r + 8*hf) lives in vgpr r at lane (16*hf + 8)
#pragma unroll
    for (int r = 0; r < 8; ++r) {
      const float li = __shfl(o[r], 16 * hf + DK);
      if (lm < DK) {
        const int m = r + 8 * hf;
        y[(size_t)(b * SS + q0 + m) * EE + h * NW + lm] = o[r] / li;
      }
    }
  }
}

// ---------------------------------------------------------------------------
// Kernel 2: out = y @ w_out^T + b_out   (8192 x 128 x 128), fp32 WMMA,
// one wave per 16x16 output tile, K=128 as 32 chained 16x16x4 WMMAs.
// ---------------------------------------------------------------------------
__global__ __launch_bounds__(256)
void proj_kernel(const float* __restrict__ y, const float* __restrict__ w,
                 const float* __restrict__ bias, float* __restrict__ out) {
  const int wid  = blockIdx.x * 8 + (threadIdx.x >> 5);  // 0..4095
  const int tm   = wid >> 3;                             // 0..511 row tile
  const int tn   = wid & 7;                              // 0..7   col tile
  const int lane = threadIdx.x & 31;
  const int hf   = lane >> 4;
  const int lm   = lane & 15;

  const float* ya = y + (size_t)(tm * 16 + lm) * EE + hf * 2;
  const float* wb = w + (size_t)(tn * 16 + lm) * EE + hf * 2;

  v8f acc = {0.f, 0.f, 0.f, 0.f, 0.f, 0.f, 0.f, 0.f};
#pragma unroll
  for (int kc = 0; kc < EE / 4; ++kc) {
    v2f a  = {ya[kc * 4], ya[kc * 4 + 1]};
    v2f bb = {wb[kc * 4], wb[kc * 4 + 1]};
    acc = __builtin_amdgcn_wmma_f32_16x16x4_f32(false, a, false, bb,
                                                (short)0, acc, false, false);
  }

  const float bo = bias[tn * 16 + lm];
#pragma unroll
  for (int r = 0; r < 8; ++r) {
    const int m = tm * 16 + r + 8 * hf;
    out[(size_t)m * EE + tn * 16 + lm] = acc[r] + bo;
  }
}

// ---------------------------------------------------------------------------
extern "C" void kernel_launch(void* const* d_in, const int* in_sizes, int n_in,
                              void* d_out, int out_size, void* d_ws, size_t ws_size,
                              hipStream_t stream) {
  const float* x     = (const float*)d_in[0];   // (8,1024,128)
  const float* theta = (const float*)d_in[1];   // (8,)
  const float* w_out = (const float*)d_in[2];   // (128,128)
  const float* b_out = (const float*)d_in[3];   // (128,)
  float* out = (float*)d_out;                   // (8,1024,128)
  float* y   = (float*)d_ws;                    // intermediate (8,1024,128) = 4MB

  qattn_kernel<<<BB * HH, 256, 0, stream>>>(x, theta, y);
  proj_kernel<<<512, 256, 0, stream>>>(y, w_out, b_out, out);
}